// Agent_72026601554520
// MI455X (gfx1250) — compile-verified
//
#include <hip/hip_runtime.h>
#include <hip/hip_bf16.h>
#include <math.h>
#include <stdint.h>

typedef __attribute__((ext_vector_type(2))) float v2f;
typedef __attribute__((ext_vector_type(8))) float v8f;

// Feature probe: if the toolchain declares the gfx1250 async-load builtin, its
// assembler also supports the instruction; we emit it via inline asm to avoid
// the builtin's CUDA-LangAS-qualified pointer parameters.
#if defined(__gfx1250__) && __has_builtin(__builtin_amdgcn_global_load_async_to_lds_b128)
#define USE_ASYNC_LDS 1
#if __has_builtin(__builtin_amdgcn_s_wait_asynccnt)
#define WAIT_ASYNC() __builtin_amdgcn_s_wait_asynccnt(0)
#else
#define WAIT_ASYNC() asm volatile("s_wait_asynccnt 0x0" ::: "memory")
#endif
#else
#define USE_ASYNC_LDS 0
#endif

#if USE_ASYNC_LDS
// Copy 16B global -> LDS asynchronously (tracked by ASYNCcnt).
// LDS byte offset = low 32 bits of the generic shared-memory address.
__device__ __forceinline__ void async_cp16(const float* gsrc, const float* ldst) {
  unsigned int lds_off = (unsigned int)(uintptr_t)ldst;
  unsigned long long gaddr = (unsigned long long)(uintptr_t)gsrc;
  asm volatile("global_load_async_to_lds_b128 %0, %1, off"
               :: "v"(lds_off), "v"(gaddr) : "memory");
}
#endif

#define N_TOT   8192
#define NUM_REL 1000
#define B_SZ    256
#define TESTT   32
#define MAX_OUT 100

// workspace layout (float offsets); all multiples of 4 -> 16B aligned
#define WS_XCAT  0                               // N*512 ; reused as state_query (h_new = cols 0..255)
#define WS_GATES (N_TOT*512)                     // N*1024
#define WS_HID   WS_GATES                        // N*512   (after gates consumed)
#define WS_MLP   (WS_GATES + N_TOT*512)          // N*256
#define WS_CNEW  (WS_GATES + N_TOT*1024)         // N*256
#define WS_TRAIL (WS_CNEW + N_TOT*256)           // B*3200
#define WS_WCAT  (WS_TRAIL + B_SZ*TESTT*MAX_OUT) // 1024*512
#define WS_BSUM  (WS_WCAT + 1024*512)            // 1024

// ---------------------------------------------------------------- prep kernels
__global__ __launch_bounds__(256) void prep_xcat(const int* __restrict__ prev_rel,
                                                 const float* __restrict__ prev_h,
                                                 const float* __restrict__ rel_emb,
                                                 float* __restrict__ xcat) {
  int i = blockIdx.x * 256 + threadIdx.x;           // N*512 threads
  int n = i >> 9, k = i & 511;
  xcat[i] = (k < 256) ? rel_emb[prev_rel[n] * 256 + k] : prev_h[n * 256 + (k - 256)];
}

__global__ __launch_bounds__(256) void prep_wcat(const float* __restrict__ W_ih,
                                                 const float* __restrict__ W_hh,
                                                 const float* __restrict__ b_ih,
                                                 const float* __restrict__ b_hh,
                                                 float* __restrict__ wcat,
                                                 float* __restrict__ bsum) {
  int i = blockIdx.x * 256 + threadIdx.x;           // 1024*512 threads
  int g = i >> 9, k = i & 511;
  wcat[i] = (k < 256) ? W_ih[g * 256 + k] : W_hh[g * 256 + k - 256];
  if (i < 1024) bsum[i] = b_ih[i] + b_hh[i];
}

// ---------------------------------------------------------------- WMMA GEMM core
// One 32-wide K-chunk of the 2x2 16x16 wave tile (fragments per ISA 7.12.2 f32 layouts).
__device__ __forceinline__ void gemm_compute_chunk(const float* As, const float* Bs,
                                                   int wr, int wc, int h, int m, v8f acc[2][2]) {
#pragma unroll
  for (int k4 = 0; k4 < 32; k4 += 4) {
    v2f a0 = *(const v2f*)&As[(wr * 32 + m) * 36 + k4 + 2 * h];
    v2f a1 = *(const v2f*)&As[(wr * 32 + 16 + m) * 36 + k4 + 2 * h];
    v2f b0 = *(const v2f*)&Bs[(wc * 32 + m) * 36 + k4 + 2 * h];
    v2f b1 = *(const v2f*)&Bs[(wc * 32 + 16 + m) * 36 + k4 + 2 * h];
    acc[0][0] = __builtin_amdgcn_wmma_f32_16x16x4_f32(false, a0, false, b0, (short)0, acc[0][0], false, false);
    acc[0][1] = __builtin_amdgcn_wmma_f32_16x16x4_f32(false, a0, false, b1, (short)0, acc[0][1], false, false);
    acc[1][0] = __builtin_amdgcn_wmma_f32_16x16x4_f32(false, a1, false, b0, (short)0, acc[1][0], false, false);
    acc[1][1] = __builtin_amdgcn_wmma_f32_16x16x4_f32(false, a1, false, b1, (short)0, acc[1][1], false, false);
  }
}

// C(M x Ncols) = act( X(M x K) @ W(Ncols x K)^T + bias ), fp32 via V_WMMA_F32_16X16X4_F32.
// Block tile 64x128, K-chunk 32, 8 waves (2x4). Staging: double-buffered async
// global->LDS copies (ASYNCcnt) so the DMA of chunk c+1 overlaps WMMA on chunk c.
__global__ __launch_bounds__(256) void gemm_wmma(const float* __restrict__ X,
                                                 const float* __restrict__ W,
                                                 const float* __restrict__ bias,
                                                 float* __restrict__ C,
                                                 int M, int Ncols, int K, int doRelu) {
  const int tid = threadIdx.x;
  const int rowBase = blockIdx.y * 64;
  const int colBase = blockIdx.x * 128;
  const int w = tid >> 5, lane = tid & 31;
  const int h = lane >> 4, m = lane & 15;  // lane-half, row/col-in-tile
  const int wr = w >> 2, wc = w & 3;       // 2x4 wave grid

  v8f acc[2][2];
  acc[0][0] = (v8f){}; acc[0][1] = (v8f){}; acc[1][0] = (v8f){}; acc[1][1] = (v8f){};

#if USE_ASYNC_LDS
  __shared__ float As[2][64 * 36];
  __shared__ float Bs[2][128 * 36];
  auto stage = [&](int kc, int p) {
#pragma unroll
    for (int it = 0; it < 2; ++it) {  // A: 512 x float4, 16B per lane per op
      int idx = tid + it * 256;
      int r = idx >> 3, kq = (idx & 7) << 2;
      async_cp16(&X[(size_t)(rowBase + r) * K + kc + kq], &As[p][r * 36 + kq]);
    }
#pragma unroll
    for (int it = 0; it < 4; ++it) {  // B: 1024 x float4
      int idx = tid + it * 256;
      int c = idx >> 3, kq = (idx & 7) << 2;
      async_cp16(&W[(size_t)(colBase + c) * K + kc + kq], &Bs[p][c * 36 + kq]);
    }
  };
  stage(0, 0);
  WAIT_ASYNC();
  __syncthreads();
  const int nCh = K >> 5;
  for (int c = 0; c < nCh; ++c) {
    int p = c & 1;
    if (c + 1 < nCh) stage((c + 1) << 5, p ^ 1);  // async prefetch next chunk
    gemm_compute_chunk(As[p], Bs[p], wr, wc, h, m, acc);
    WAIT_ASYNC();
    __syncthreads();
  }
#else
  __shared__ float As[64 * 36];
  __shared__ float Bs[128 * 36];
  for (int kc = 0; kc < K; kc += 32) {
#pragma unroll
    for (int it = 0; it < 2; ++it) {
      int idx = tid + it * 256;
      int r = idx >> 3, kq = (idx & 7) << 2;
      float4 v = *(const float4*)&X[(size_t)(rowBase + r) * K + kc + kq];
      *(float4*)&As[r * 36 + kq] = v;
    }
#pragma unroll
    for (int it = 0; it < 4; ++it) {
      int idx = tid + it * 256;
      int c = idx >> 3, kq = (idx & 7) << 2;
      float4 v = *(const float4*)&W[(size_t)(colBase + c) * K + kc + kq];
      *(float4*)&Bs[c * 36 + kq] = v;
    }
    __syncthreads();
    gemm_compute_chunk(As, Bs, wr, wc, h, m, acc);
    __syncthreads();
  }
#endif

#pragma unroll
  for (int i = 0; i < 2; ++i)
#pragma unroll
    for (int j = 0; j < 2; ++j) {
      int col = colBase + wc * 32 + j * 16 + m;
      float bv = bias[col];
#pragma unroll
      for (int r = 0; r < 8; ++r) {  // D: lane m = N, M = r + 8*h
        int row = rowBase + wr * 32 + i * 16 + 8 * h + r;
        float v = acc[i][j][r] + bv;
        if (doRelu) v = fmaxf(v, 0.0f);
        C[(size_t)row * Ncols + col] = v;
      }
    }
}

// ---------------------------------------------------------------- LSTM pointwise + state_query build
__global__ __launch_bounds__(256) void lstm_pointwise(const float* __restrict__ gates,
                                                      const float* __restrict__ prev_c,
                                                      const int* __restrict__ queries,
                                                      const float* __restrict__ rel_emb,
                                                      float* __restrict__ cnew,
                                                      float* __restrict__ sq) {
  int i = blockIdx.x * 256 + threadIdx.x;  // N*256 threads
  int n = i >> 8, s = i & 255;
  float gi = gates[n * 1024 + s];
  float gf = gates[n * 1024 + 256 + s];
  float gg = gates[n * 1024 + 512 + s];
  float go = gates[n * 1024 + 768 + s];
  float c = prev_c[i];
  float si = 1.0f / (1.0f + expf(-gi));
  float sf = 1.0f / (1.0f + expf(-gf));
  float so = 1.0f / (1.0f + expf(-go));
  float cn = sf * c + si * tanhf(gg);
  float hn = so * tanhf(cn);
  cnew[i] = cn;
  sq[n * 512 + s] = hn;                              // h_new kept for final beam gather
  sq[n * 512 + 256 + s] = rel_emb[queries[n] * 256 + s];
}

// ---------------------------------------------------------------- fused scores + log_softmax + trail
// 1 block per 16 rows: WMMA scores vs all 1000 relations into LDS, gather by action id,
// mask PAD, wave-shuffle log-softmax over MAX_OUT, add log_current_prob, emit trail.
__global__ __launch_bounds__(256) void score_softmax(const float* __restrict__ mlp,
                                                     const float* __restrict__ rel_emb,
                                                     const int* __restrict__ actions,
                                                     const float* __restrict__ logcur,
                                                     float* __restrict__ trail) {
  __shared__ float sc[16 * 1008];  // 63 KB, fits even a 64KB LDS cap
  const int tid = threadIdx.x;
  const int n0 = blockIdx.x * 16;
  const int w = tid >> 5, lane = tid & 31;
  const int h = lane >> 4, m = lane & 15;
  const float* Arow = mlp + (size_t)(n0 + m) * 256;

  for (int t = w; t < 63; t += 8) {  // 63 rel tiles of 16 (1008, cols >=1000 clamped, never gathered)
    int colb = t * 16;
    int rel = colb + m; if (rel >= NUM_REL) rel = NUM_REL - 1;
    const float* Brow = rel_emb + (size_t)rel * 256;
    v8f acc = (v8f){};
#pragma unroll 4
    for (int k = 0; k < 256; k += 4) {
      v2f a = *(const v2f*)&Arow[k + 2 * h];
      v2f b = *(const v2f*)&Brow[k + 2 * h];
      acc = __builtin_amdgcn_wmma_f32_16x16x4_f32(false, a, false, b, (short)0, acc, false, false);
    }
#pragma unroll
    for (int r = 0; r < 8; ++r) sc[(r + 8 * h) * 1008 + colb + m] = acc[r];
  }
  __syncthreads();

  for (int nl = w; nl < 16; nl += 8) {  // each wave: rows w and w+8
    int n = n0 + nl;
    float lc = logcur[n];
    float v[4];
#pragma unroll
    for (int q = 0; q < 4; ++q) {
      int mm = lane + q * 32;
      if (mm < MAX_OUT) {
        int rid = actions[((size_t)n * MAX_OUT + mm) * 2];
        v[q] = (rid == 0) ? -99999.0f : sc[nl * 1008 + rid];
      } else v[q] = -INFINITY;
    }
    float mx = fmaxf(fmaxf(v[0], v[1]), fmaxf(v[2], v[3]));
    for (int off = 16; off > 0; off >>= 1) mx = fmaxf(mx, __shfl_xor(mx, off, 32));
    float s = 0.0f;
#pragma unroll
    for (int q = 0; q < 4; ++q) s += expf(v[q] - mx);
    for (int off = 16; off > 0; off >>= 1) s += __shfl_xor(s, off, 32);
    float lse = mx + logf(s);
#pragma unroll
    for (int q = 0; q < 4; ++q) {
      int mm = lane + q * 32;
      if (mm < MAX_OUT) trail[(size_t)n * MAX_OUT + mm] = v[q] - lse + lc;
    }
  }
}

// ---------------------------------------------------------------- top-32 per batch row + output gathers
__global__ __launch_bounds__(256) void topk_gather(const float* __restrict__ trail,
                                                   const int* __restrict__ actions,
                                                   const float* __restrict__ sq,
                                                   const float* __restrict__ cnew,
                                                   float* __restrict__ outH, float* __restrict__ outC,
                                                   float* __restrict__ outRel, float* __restrict__ outEnt,
                                                   float* __restrict__ outLog) {
  __shared__ float tv[3200];
  __shared__ float rv[256];
  __shared__ int ri[256];
  __shared__ int selIdx[32];
  __shared__ float selVal[32];
  const int b = blockIdx.x;
  const int tid = threadIdx.x;
  for (int i = tid; i < 3200; i += 256) tv[i] = trail[(size_t)b * 3200 + i];
  __syncthreads();
  for (int t = 0; t < 32; ++t) {
    float best = -INFINITY; int bi = 0x7fffffff;
    for (int i = tid; i < 3200; i += 256) {
      float v = tv[i];
      if (v > best) { best = v; bi = i; }   // ascending scan -> lowest index on ties
    }
    rv[tid] = best; ri[tid] = bi;
    __syncthreads();
    for (int s = 128; s > 0; s >>= 1) {
      if (tid < s) {
        float ov = rv[tid + s]; int oi = ri[tid + s];
        if (ov > rv[tid] || (ov == rv[tid] && oi < ri[tid])) { rv[tid] = ov; ri[tid] = oi; }
      }
      __syncthreads();
    }
    if (tid == 0) { selIdx[t] = ri[0]; selVal[t] = rv[0]; tv[ri[0]] = -INFINITY; }
    __syncthreads();
  }
  if (tid < 32) {
    int t = tid, p = selIdx[t];
    int beam = p / MAX_OUT, mm = p - beam * MAX_OUT;
    int nsrc = b * TESTT + beam;
    int ndst = b * TESTT + t;
    outRel[ndst] = (float)actions[((size_t)nsrc * MAX_OUT + mm) * 2 + 0];
    outEnt[ndst] = (float)actions[((size_t)nsrc * MAX_OUT + mm) * 2 + 1];
    outLog[ndst] = selVal[t];
  }
  __syncthreads();
  for (int t = 0; t < 32; ++t) {  // beam gathers of h_new (lives in sq cols 0..255) and c_new
    int beam = selIdx[t] / MAX_OUT;
    int nsrc = b * TESTT + beam;
    int ndst = b * TESTT + t;
    outH[(size_t)ndst * 256 + tid] = sq[(size_t)nsrc * 512 + tid];
    outC[(size_t)ndst * 256 + tid] = cnew[(size_t)nsrc * 256 + tid];
  }
}

// ---------------------------------------------------------------- launch
extern "C" void kernel_launch(void* const* d_in, const int* in_sizes, int n_in,
                              void* d_out, int out_size, void* d_ws, size_t ws_size,
                              hipStream_t stream) {
  (void)in_sizes; (void)n_in; (void)out_size; (void)ws_size;
  const float* prev_h   = (const float*)d_in[0];
  const float* prev_c   = (const float*)d_in[1];
  const int*   prev_rel = (const int*)d_in[2];
  const int*   actions  = (const int*)d_in[3];
  const int*   queries  = (const int*)d_in[4];
  const float* logcur   = (const float*)d_in[5];
  const float* rel_emb  = (const float*)d_in[7];
  const float* W_ih     = (const float*)d_in[8];
  const float* W_hh     = (const float*)d_in[9];
  const float* b_ih     = (const float*)d_in[10];
  const float* b_hh     = (const float*)d_in[11];
  const float* W1       = (const float*)d_in[12];
  const float* b1       = (const float*)d_in[13];
  const float* W2       = (const float*)d_in[14];
  const float* b2       = (const float*)d_in[15];

  float* ws    = (float*)d_ws;
  float* xcat  = ws + WS_XCAT;   // later: state_query
  float* gates = ws + WS_GATES;
  float* hid   = ws + WS_HID;
  float* mlp   = ws + WS_MLP;
  float* cnew  = ws + WS_CNEW;
  float* trail = ws + WS_TRAIL;
  float* wcat  = ws + WS_WCAT;
  float* bsum  = ws + WS_BSUM;

  float* out    = (float*)d_out;
  float* outH   = out;
  float* outC   = out + (size_t)N_TOT * 256;
  float* outRel = out + (size_t)2 * N_TOT * 256;
  float* outEnt = outRel + N_TOT;
  float* outLog = outEnt + N_TOT;

  prep_xcat<<<N_TOT * 512 / 256, 256, 0, stream>>>(prev_rel, prev_h, rel_emb, xcat);
  prep_wcat<<<1024 * 512 / 256, 256, 0, stream>>>(W_ih, W_hh, b_ih, b_hh, wcat, bsum);
  gemm_wmma<<<dim3(1024 / 128, N_TOT / 64), 256, 0, stream>>>(xcat, wcat, bsum, gates, N_TOT, 1024, 512, 0);
  lstm_pointwise<<<N_TOT * 256 / 256, 256, 0, stream>>>(gates, prev_c, queries, rel_emb, cnew, xcat);
  gemm_wmma<<<dim3(512 / 128, N_TOT / 64), 256, 0, stream>>>(xcat, W1, b1, hid, N_TOT, 512, 512, 1);
  gemm_wmma<<<dim3(256 / 128, N_TOT / 64), 256, 0, stream>>>(hid, W2, b2, mlp, N_TOT, 256, 512, 1);
  score_softmax<<<N_TOT / 16, 256, 0, stream>>>(mlp, rel_emb, actions, logcur, trail);
  topk_gather<<<B_SZ, 256, 0, stream>>>(trail, actions, xcat, cnew, outH, outC, outRel, outEnt, outLog);
}